// MIModel_16698832846897
// MI455X (gfx1250) — compile-verified
//
#include <hip/hip_runtime.h>
#include <math.h>

typedef float v2f __attribute__((ext_vector_type(2)));
typedef float v8f __attribute__((ext_vector_type(8)));

#define D 128
#define G 384          // 3*D GRU gate rows
#define THRESH 0.1f

struct Smem {
  float E[3][D];       // embedding rows e0,e1,e2
  float X[3][D];       // surviving intents (zero-padded to 3)
  float GI[3][G];      // Wih @ X[t]
  float GH[G];         // Whh @ h
  float h[D];          // GRU hidden state
  float HS[3][D];      // GRU outputs per step
  float dots[6];       // d01,d02,d12, w0,w1,w2
  float rel[3];        // attention logits
};

__device__ __forceinline__ float sigf(float x) { return 1.0f / (1.0f + __expf(-x)); }

__device__ __forceinline__ void softmax2(float a, float b, float* pa, float* pb) {
  float m = fmaxf(a, b);
  float ea = __expf(a - m), eb = __expf(b - m);
  float inv = 1.0f / (ea + eb);
  *pa = ea * inv; *pb = eb * inv;
}

// dot of two 128-float vectors by one wave32; result valid in lane 0
__device__ __forceinline__ float wave_dot128(const float* a, const float* b, int lane) {
  float s = 0.0f;
  #pragma unroll
  for (int i = 0; i < 4; ++i) s += a[lane + 32 * i] * b[lane + 32 * i];
  #pragma unroll
  for (int off = 16; off > 0; off >>= 1) s += __shfl_down(s, off, 32);
  return s;
}

// Out[t*G + m] = sum_k M[m,k] * Bsrc[t*D + k]   for m in [0,384), t in [0,ncols)
// M: global 384x128 row-major. Uses V_WMMA_F32_16X16X4_F32, 8 waves x 3 M-tiles.
// A layout (ISA 7.12.2, 32-bit 16x4): lane half selects K pair, reg selects K in pair.
// B mirrors A transposed: reg r, lane -> B[2*(lane>>4)+r][lane&15].
// D (16x16 f32): reg v, lanes0-15 -> M=v, lanes16-31 -> M=v+8, N=lane&15.
__device__ void wmma_matT(const float* __restrict__ M, const float* Bsrc,
                          float* Out, int ncols, int tid) {
  const int lane = tid & 31;
  const int wave = tid >> 5;
  const int half = lane >> 4;
  const int l16  = lane & 15;
  const int  bcol  = (l16 < ncols) ? l16 : 0;     // clamp + mask: no EXEC change
  const float bmsk = (l16 < ncols) ? 1.0f : 0.0f;
  for (int tile = 0; tile < 3; ++tile) {
    const int m0 = (wave * 3 + tile) * 16;
    v8f c = {0.f, 0.f, 0.f, 0.f, 0.f, 0.f, 0.f, 0.f};
    const float* mrow = M + (size_t)(m0 + l16) * D;
    for (int k = 0; k < D; k += 4) {
      v2f a, b;
      a.x = mrow[k + half * 2 + 0];
      a.y = mrow[k + half * 2 + 1];
      b.x = bmsk * Bsrc[bcol * D + k + half * 2 + 0];
      b.y = bmsk * Bsrc[bcol * D + k + half * 2 + 1];
      c = __builtin_amdgcn_wmma_f32_16x16x4_f32(false, a, false, b, (short)0, c,
                                                false, false);
    }
    if (l16 < ncols) {
      const int rbase = m0 + half * 8;
      #pragma unroll
      for (int v = 0; v < 8; ++v) Out[l16 * G + rbase + v] = c[v];
    }
  }
}

__global__ __launch_bounds__(256)
void mimodel_fused_kernel(const float* __restrict__ emb, const float* __restrict__ W,
                          const float* __restrict__ attW, const float* __restrict__ Wih,
                          const float* __restrict__ Whh, const float* __restrict__ bih,
                          const float* __restrict__ bhh, const int* __restrict__ sess,
                          float* __restrict__ out) {
  __shared__ Smem s;
  const int tid  = threadIdx.x;
  const int lane = tid & 31;
  const int wave = tid >> 5;

  // --- load the 3 session embedding rows ---
  const int id0 = sess[0], id1 = sess[1], id2 = sess[2];
  for (int i = tid; i < 3 * D; i += 256) {
    const int r = i >> 7, c0 = i & (D - 1);
    const int id = (r == 0) ? id0 : (r == 1) ? id1 : id2;
    s.E[r][c0] = emb[(size_t)id * D + c0];
  }
  __syncthreads();

  // --- 6 primitive dot products, one wave each ---
  if (wave < 6) {
    const float *a, *b;
    switch (wave) {
      case 0: a = s.E[0]; b = s.E[1]; break;
      case 1: a = s.E[0]; b = s.E[2]; break;
      case 2: a = s.E[1]; b = s.E[2]; break;
      case 3: a = s.E[0]; b = W;      break;
      case 4: a = s.E[1]; b = W;      break;
      default:a = s.E[2]; b = W;      break;
    }
    float r = wave_dot128(a, b, lane);
    if (lane == 0) s.dots[wave] = r;
  }
  __syncthreads();

  // --- uniform scalar control flow (every thread computes identically) ---
  const float d01 = s.dots[0], d02 = s.dots[1], d12 = s.dots[2];
  const float w0  = s.dots[3], w1  = s.dots[4], w2  = s.dots[5];
  const bool take1 = sigf(d01) > THRESH;
  const bool ta    = sigf(d02) > THRESH;
  const bool tb    = sigf(d12) > THRESH;
  float p0, p1, q0, q1, r0, r1, u0, u1;
  softmax2(w0, w1, &p0, &p1);                 // u01 weights
  softmax2(w0, w2, &q0, &q1);                 // u02 weights
  softmax2(w1, w2, &r0, &r1);                 // u12 weights
  const float wu = p0 * w0 + p1 * w1;         // dot(u01, W), analytic
  const float du = p0 * d02 + p1 * d12;       // dot(u01, e2), analytic
  const bool tc = take1 && ta && tb && (sigf(du) > THRESH);
  softmax2(wu, w2, &u0, &u1);                 // u012 weights

  // candidate intents as affine combos of (e0,e1,e2), in dict insertion order
  const float C[7][3] = {
    {1.f, 0.f, 0.f}, {0.f, 1.f, 0.f}, {p0, p1, 0.f}, {0.f, 0.f, 1.f},
    {q0, 0.f, q1},   {0.f, r0, r1},   {u0 * p0, u0 * p1, u1}
  };
  bool keep[7];
  if (take1) {
    keep[0] = false;       keep[1] = false;       keep[2] = !tc;
    keep[3] = !(ta || tb); keep[4] = ta && !tc;   keep[5] = tb && !tc;
    keep[6] = tc;          // subsumption prunes (0,2),(1,2) when (0,1,2) exists
  } else {
    keep[0] = !ta;         keep[1] = !tb;         keep[2] = false;
    keep[3] = !(ta || tb); keep[4] = ta;          keep[5] = tb;
    keep[6] = false;
  }
  int N = 0;
  #pragma unroll
  for (int cnd = 0; cnd < 7; ++cnd) {
    if (keep[cnd]) {
      if (tid < D)
        s.X[N][tid] = C[cnd][0] * s.E[0][tid] + C[cnd][1] * s.E[1][tid] +
                      C[cnd][2] * s.E[2][tid];
      N++;                                   // N <= 3 by construction
    }
  }
  for (int r = N; r < 3; ++r) if (tid < D) s.X[r][tid] = 0.0f;
  if (tid < D) s.h[tid] = 0.0f;
  __syncthreads();

  // --- GI = Wih @ X^T : all GRU input gates in one f32 WMMA GEMM ---
  wmma_matT(Wih, &s.X[0][0], &s.GI[0][0], 3, tid);
  __syncthreads();

  // --- GRU recurrence (torch gate order r,z,n), N steps ---
  for (int t = 0; t < N; ++t) {
    wmma_matT(Whh, s.h, s.GH, 1, tid);       // GH = Whh @ h
    __syncthreads();
    if (tid < D) {
      const float gr  = s.GI[t][tid]         + bih[tid]         + s.GH[tid]     + bhh[tid];
      const float gz  = s.GI[t][D + tid]     + bih[D + tid]     + s.GH[D + tid] + bhh[D + tid];
      const float gin = s.GI[t][2 * D + tid] + bih[2 * D + tid];
      const float ghn = s.GH[2 * D + tid]    + bhh[2 * D + tid];
      const float rr = sigf(gr);
      const float zz = sigf(gz);
      const float nn = tanhf(gin + rr * ghn);
      const float hn = (1.0f - zz) * nn + zz * s.h[tid];
      s.h[tid]     = hn;
      s.HS[t][tid] = hn;
    }
    __syncthreads();
  }

  // --- attention readout: rel_t = hs_t . attW, softmax over t, weighted sum ---
  if (wave < N) {
    float r = wave_dot128(s.HS[wave], attW, lane);
    if (lane == 0) s.rel[wave] = r;
  }
  __syncthreads();

  if (tid < D) {
    float m = -3.402823e38f;
    for (int t = 0; t < N; ++t) m = fmaxf(m, s.rel[t]);
    float e[3], esum = 0.0f;
    for (int t = 0; t < N; ++t) { e[t] = __expf(s.rel[t] - m); esum += e[t]; }
    float acc = 0.0f;
    for (int t = 0; t < N; ++t) acc += (e[t] / esum) * s.HS[t][tid];
    out[tid] = acc;
  }
}

extern "C" void kernel_launch(void* const* d_in, const int* in_sizes, int n_in,
                              void* d_out, int out_size, void* d_ws, size_t ws_size,
                              hipStream_t stream) {
  const float* emb  = (const float*)d_in[0];
  const float* W    = (const float*)d_in[1];
  const float* attW = (const float*)d_in[2];
  const float* Wih  = (const float*)d_in[3];
  const float* Whh  = (const float*)d_in[4];
  const float* bih  = (const float*)d_in[5];
  const float* bhh  = (const float*)d_in[6];
  const int*   sess = (const int*)d_in[7];
  float* out = (float*)d_out;
  mimodel_fused_kernel<<<1, 256, 0, stream>>>(emb, W, attW, Wih, Whh, bih, bhh,
                                              sess, out);
}